// EAConv_7361573945622
// MI455X (gfx1250) — compile-verified
//
#include <hip/hip_runtime.h>
#include <cstdint>
#include <math.h>

// Problem constants (from reference)
#define T_STEPS 8
#define NN      20000
#define DD      128
#define MM      32
#define KKC     8
#define DEL     16
#define ITERS   5
#define AGGP    0.5f

// LDS layout strides (conflict avoidance)
#define ZSTRIDE 132   // 128 + 4 pad: row stride in floats -> rows land on different banks
#define PSTRIDE 9     // 8 + 1 pad

typedef float v2f __attribute__((ext_vector_type(2)));
typedef float v8f __attribute__((ext_vector_type(8)));

__device__ __forceinline__ void lds_fence() { __asm__ volatile("" ::: "memory"); }

// ---------------------------------------------------------------------------
// Kernel 1: per-factor L2 normalize, write xn table of N+1 rows (row N = zeros)
// ---------------------------------------------------------------------------
__global__ void eac_normalize_kernel(const float* __restrict__ x, float* __restrict__ xn) {
  int idx = blockIdx.x * blockDim.x + threadIdx.x;
  if (idx >= (NN + 1) * KKC) return;
  int n = idx >> 3;
  int k = idx & 7;
  float* dst = xn + (long)n * DD + k * DEL;
  if (n < NN) {
    const float* src = x + (long)n * DD + k * DEL;
    float v[DEL];
#pragma unroll
    for (int i = 0; i < DEL; ++i) v[i] = src[i];
    float ss = 0.f;
#pragma unroll
    for (int i = 0; i < DEL; ++i) ss += v[i] * v[i];
    float sc = 1.f / fmaxf(sqrtf(ss), 1e-12f);
#pragma unroll
    for (int i = 0; i < DEL; ++i) dst[i] = v[i] * sc;
  } else {
#pragma unroll
    for (int i = 0; i < DEL; ++i) dst[i] = 0.f;
  }
}

// ---------------------------------------------------------------------------
// Kernel 2: one wave per node. Async-gather z tile into LDS, 5 routing
// iterations (p-einsum via v_wmma_f32_16x16x4_f32 chain, u-einsum VALU),
// fused temporal-scan epilogue.
// ---------------------------------------------------------------------------
__global__ __launch_bounds__(64, 1)
void eac_route_kernel(const float* __restrict__ xn, const int* __restrict__ nb,
                      float* __restrict__ out_t, float* __restrict__ S,
                      int t, float sig_t, float inv_t) {
  __shared__ float z_sh[2][MM * ZSTRIDE];   // 2 x 16.5KB : z tiles (f32)
  __shared__ float p_sh[2][MM * PSTRIDE];   // 2 x 1.1KB  : routing weights
  __shared__ float u_sh[2][DD];             // 2 x 0.5KB  : current u

  const int lane = threadIdx.x & 31;
  const int wid  = threadIdx.x >> 5;
  const int n    = blockIdx.x * 2 + wid;
  if (n >= NN) return;                       // uniform per wave

  float* z  = z_sh[wid];
  float* ps = p_sh[wid];
  float* us = u_sh[wid];

  // ---- gather: lane m pulls the first 128B of neighbor row m into LDS, async path ----
  {
    int nbr = nb[(long)n * MM + lane];
    unsigned long long ga = (unsigned long long)(uintptr_t)(xn + (long)nbr * DD);
    unsigned int la = (unsigned int)(uintptr_t)&z[lane * ZSTRIDE];
#pragma unroll
    for (int j = 0; j < 8; ++j) {
      unsigned int laj = la + (unsigned int)(j * 16);
      unsigned long long gaj = ga + (unsigned long long)(j * 16);
      asm volatile("global_load_async_to_lds_b128 %0, %1, off"
                   :: "v"(laj), "v"(gaj)
                   : "memory");
    }
  }

  // self factors: lane holds x_fac for (k = lane>>2, dquad = lane&3) == dims 4*lane..4*lane+3
  float4 xs = *(const float4*)(xn + (long)n * DD + lane * 4);

  asm volatile("s_wait_asynccnt 0" ::: "memory");
  lds_fence();

  const int c  = lane & 15;   // WMMA column / A-tile row index
  const int h  = lane >> 4;   // WMMA lane-half
  const int kk = lane >> 2;   // capsule owned by this lane in the u-einsum

  float ux, uy, uz, uw;

  // ---- iteration 0: p = softmax(0) = 1/8 uniform; u = (1/8) sum_m z + x_fac ----
  {
    float ax = xs.x, ay = xs.y, az = xs.z, aw = xs.w;
#pragma unroll 8
    for (int m2 = 0; m2 < MM; ++m2) {
      float4 z4 = *(const float4*)&z[m2 * ZSTRIDE + lane * 4];
      ax += 0.125f * z4.x; ay += 0.125f * z4.y; az += 0.125f * z4.z; aw += 0.125f * z4.w;
    }
    float ssq = ax * ax + ay * ay + az * az + aw * aw;
    ssq += __shfl_xor(ssq, 1, 32);
    ssq += __shfl_xor(ssq, 2, 32);
    float sc = 1.f / fmaxf(sqrtf(ssq), 1e-12f);
    ax *= sc; ay *= sc; az *= sc; aw *= sc;
    *(float4*)&us[lane * 4] = make_float4(ax, ay, az, aw);
    lds_fence();
    ux = ax; uy = ay; uz = az; uw = aw;   // (overwritten below)
  }

#pragma unroll 1
  for (int it = 1; it < ITERS; ++it) {
    // ---- preload this lane's 8 possible B values (unconditional ds_b64 x4) ----
    // lane needs u[c][4q + 2h + {0,1}] for q = 0..3; lanes with c>=8 select 0 later.
    v2f ur[4];
#pragma unroll
    for (int q = 0; q < 4; ++q)
      ur[q] = *(const v2f*)&us[((c & 7) << 4) + (q << 2) + (h << 1)];

    // ---- p-einsum: P(32x16) = Z(32x128) * Ublk(128x16), block-diag Ublk ----
    // chain of v_wmma_f32_16x16x4_f32 over the 128-deep contraction, 32 steps
    v8f acc0 = {};
    v8f acc1 = {};
#pragma unroll
    for (int s = 0; s < 32; ++s) {
      const int ks = s >> 2;                 // capsule of this d-chunk
      const int q  = s & 3;
      const bool act = (c == ks);
      v2f b;
      b.x = act ? ur[q].x : 0.f;             // B[2h,   c]
      b.y = act ? ur[q].y : 0.f;             // B[2h+1, c]
      v2f a0 = *(const v2f*)&z[c * ZSTRIDE        + (s << 2) + (h << 1)];
      v2f a1 = *(const v2f*)&z[(16 + c) * ZSTRIDE + (s << 2) + (h << 1)];
      acc0 = __builtin_amdgcn_wmma_f32_16x16x4_f32(false, a0, false, b, (short)0, acc0, false, false);
      acc1 = __builtin_amdgcn_wmma_f32_16x16x4_f32(false, a1, false, b, (short)0, acc1, false, false);
    }

    // scatter the 8 useful columns of each 16x16 D tile to LDS
    if (c < KKC) {
#pragma unroll
      for (int i = 0; i < 8; ++i) {
        ps[(i + (h << 3)) * PSTRIDE + c]      = acc0[i];
        ps[(16 + i + (h << 3)) * PSTRIDE + c] = acc1[i];
      }
    }
    lds_fence();

    // ---- softmax over k; lane handles row m = lane ----
    {
      float v0[KKC];
#pragma unroll
      for (int i = 0; i < KKC; ++i) v0[i] = ps[lane * PSTRIDE + i];
      float mx = v0[0];
#pragma unroll
      for (int i = 1; i < KKC; ++i) mx = fmaxf(mx, v0[i]);
      float sm = 0.f;
#pragma unroll
      for (int i = 0; i < KKC; ++i) { v0[i] = __expf(v0[i] - mx); sm += v0[i]; }
      float rs = 1.f / sm;
#pragma unroll
      for (int i = 0; i < KKC; ++i) ps[lane * PSTRIDE + i] = v0[i] * rs;
    }
    lds_fence();

    // ---- u-einsum: lane accumulates u[kk][4q..4q+3] over m, plus x_fac ----
    float ax = xs.x, ay = xs.y, az = xs.z, aw = xs.w;
#pragma unroll 8
    for (int m2 = 0; m2 < MM; ++m2) {
      float w = ps[m2 * PSTRIDE + kk];
      float4 z4 = *(const float4*)&z[m2 * ZSTRIDE + lane * 4];
      ax += w * z4.x; ay += w * z4.y; az += w * z4.z; aw += w * z4.w;
    }

    if (it < ITERS - 1) {
      // per-factor normalize: reduce ||.||^2 across the 4 lanes sharing k
      float ssq = ax * ax + ay * ay + az * az + aw * aw;
      ssq += __shfl_xor(ssq, 1, 32);
      ssq += __shfl_xor(ssq, 2, 32);
      float sc = 1.f / fmaxf(sqrtf(ssq), 1e-12f);
      ax *= sc; ay *= sc; az *= sc; aw *= sc;
      *(float4*)&us[lane * 4] = make_float4(ax, ay, az, aw);
      lds_fence();
    }
    ux = ax; uy = ay; uz = az; uw = aw;   // last iteration leaves raw u = x_temp
  }

  // ---- fused temporal scan: emb_t and S update ----
  long off = (long)n * DD + lane * 4;
  float4 e;
  if (t == 0) {
    e = make_float4(ux, uy, uz, uw);
    float4 s4 = make_float4(sig_t * e.x, sig_t * e.y, sig_t * e.z, sig_t * e.w);
    *(float4*)(S + off) = s4;
  } else {
    float4 s4 = *(const float4*)(S + off);
    const float a2 = inv_t * AGGP;
    const float b2 = 1.f - AGGP;
    e.x = s4.x * a2 + ux * b2;
    e.y = s4.y * a2 + uy * b2;
    e.z = s4.z * a2 + uz * b2;
    e.w = s4.w * a2 + uw * b2;
    s4.x += sig_t * e.x; s4.y += sig_t * e.y; s4.z += sig_t * e.z; s4.w += sig_t * e.w;
    *(float4*)(S + off) = s4;
  }
  *(float4*)(out_t + off) = e;
}

// ---------------------------------------------------------------------------
extern "C" void kernel_launch(void* const* d_in, const int* in_sizes, int n_in,
                              void* d_out, int out_size, void* d_ws, size_t ws_size,
                              hipStream_t stream) {
  const float* x_all  = (const float*)d_in[0];   // (T, N, D) f32
  const int*   nb_all = (const int*)d_in[1];     // (T, N, M) i32
  (void)in_sizes; (void)n_in; (void)out_size; (void)ws_size;

  float* out = (float*)d_out;                    // (T, N, D) f32
  float* xn  = (float*)d_ws;                     // (N+1, D) normalized table
  float* S   = xn + (size_t)(NN + 1) * DD;       // (N, D) running sum

  const int norm_grid  = ((NN + 1) * KKC + 255) / 256;
  const int route_grid = (NN + 1) / 2;           // 2 nodes (waves) per block

  for (int t = 0; t < T_STEPS; ++t) {
    eac_normalize_kernel<<<norm_grid, 256, 0, stream>>>(
        x_all + (size_t)t * NN * DD, xn);

    float tf    = (float)t;
    float sig   = 1.f / (1.f + expf(-tf));
    float invt  = 1.f / fmaxf(tf, 1.f);
    eac_route_kernel<<<route_grid, 64, 0, stream>>>(
        xn, nb_all + (size_t)t * NN * MM,
        out + (size_t)t * NN * DD, S, t, sig, invt);
  }
}